// PosEmbeddingwithMask_11003706212368
// MI455X (gfx1250) — compile-verified
//
#include <hip/hip_runtime.h>

typedef __attribute__((ext_vector_type(2))) float        v2f;
typedef __attribute__((ext_vector_type(4))) float        v4f;
typedef __attribute__((ext_vector_type(8))) float        v8f;
typedef __attribute__((ext_vector_type(4))) unsigned int v4u;
typedef __attribute__((ext_vector_type(4))) int          v4i;
typedef __attribute__((ext_vector_type(8))) int          v8i;

#define L 128

#if __has_builtin(__builtin_amdgcn_tensor_load_to_lds)
#define HAVE_TDM 1
#else
#define HAVE_TDM 0
#endif

#if HAVE_TDM
__device__ __forceinline__ void tdm_load_tile16x16(
    unsigned ldsOff, const float* gptr, unsigned rowsTotal)
{
    // 16x16 f32 tile from a row-major [rowsTotal, L] tensor, tile origin = gptr.
    // LDS padding: +1 dword after every 16 dwords -> row stride 17.
    unsigned long long ga = (unsigned long long)(const void*)gptr;
    v4u g0;
    g0[0] = 1u;                                    // count=1, user descriptor
    g0[1] = ldsOff;                                // lds_addr (bytes)
    g0[2] = (unsigned)ga;                          // global_addr[31:0]
    g0[3] = (unsigned)(ga >> 32) | (2u << 30);     // global_addr[56:32] | type=2
    v8i g1;
    g1[0] = (int)((2u << 16) | (1u << 20) | (3u << 22)); // data_size=4B, pad_en, interval=16dw (pad=1dw)
    g1[1] = (int)((unsigned)L << 16);              // tensor_dim0 = 128
    g1[2] = (int)((rowsTotal & 0xFFFFu) << 16);    // tensor_dim1 lo16
    g1[3] = (int)((rowsTotal >> 16) | (16u << 16));// tensor_dim1 hi16 | tile_dim0=16
    g1[4] = 16;                                    // tile_dim1 = 16
    g1[5] = L;                                     // tensor_dim0_stride = 128
    g1[6] = 0;
    g1[7] = 0;
    v4i z4 = {0, 0, 0, 0};
#if defined(__clang_major__) && (__clang_major__ >= 23)
    v8i z8 = {0, 0, 0, 0, 0, 0, 0, 0};
    __builtin_amdgcn_tensor_load_to_lds(g0, g1, z4, z4, z8, 0);
#else
    __builtin_amdgcn_tensor_load_to_lds(g0, g1, z4, z4, 0);
#endif
}
#endif

// -------------------------------------------------------------------------
// Edge path: out[e,:] = edge_attr[e,:] + relu(len[e]*w1 + b1) @ W2 + b2
// Block owns one 16-wide feature column (nt = blockIdx&7); its 8 waves
// grid-stride over 16-edge tiles. B fragments + layer-1 coeffs live in LDS
// in WMMA-fragment order; edge_attr tiles staged by TDM, overlapped with
// the 32 x v_wmma_f32_16x16x4_f32 K-loop.
// -------------------------------------------------------------------------
__global__ void edge_mlp_wmma(
    const float* __restrict__ pos,
    const int*   __restrict__ edge_index,   // [2,E]: rows then cols
    const float* __restrict__ edge_attr,    // [E,L]
    const float* __restrict__ last_pred,
    const unsigned char* __restrict__ mask, // [N] bool
    const float* __restrict__ Wd1,          // [L]
    const float* __restrict__ bd1,          // [L]
    const float* __restrict__ Wd2,          // [L,L]
    const float* __restrict__ bd2,          // [L]
    float* __restrict__ out,                // [E,L]
    int E, int numTiles)
{
    __shared__ v2f   sB[32 * 32];        // Wd2 col-block, fragment order (8KB)
    __shared__ v4f   sC[64];             // {w1[k0],w1[k0+1],b1[k0],b1[k0+1]} per (kk,hi)
    __shared__ float sTile[8][16 * 17];  // TDM-staged edge_attr tiles, row stride 17

    const int nt = blockIdx.x & 7;

    for (int i = threadIdx.x; i < 32 * 32; i += blockDim.x) {
        const int lane_ = i & 31, kk_ = i >> 5;
        const int k0_ = kk_ * 4 + ((lane_ < 16) ? 0 : 2);
        const int n_  = nt * 16 + (lane_ & 15);
        v2f b; b.x = Wd2[k0_ * L + n_]; b.y = Wd2[(k0_ + 1) * L + n_];
        sB[i] = b;
    }
    for (int i = threadIdx.x; i < 64; i += blockDim.x) {
        const int k0_ = (i >> 1) * 4 + (i & 1) * 2;
        v4f c; c.x = Wd1[k0_]; c.y = Wd1[k0_ + 1]; c.z = bd1[k0_]; c.w = bd1[k0_ + 1];
        sC[i] = c;
    }
    __syncthreads();

    const int lane        = threadIdx.x & 31;
    const int waveInBlk   = threadIdx.x >> 5;
    const int wavesPerBlk = blockDim.x >> 5;
    const int tile0       = (blockIdx.x >> 3) * wavesPerBlk + waveInBlk;
    const int tileStride  = (gridDim.x >> 3) * wavesPerBlk;

    const int nl   = lane & 15;
    const int hi   = lane >> 4;
    const int n    = nt * 16 + nl;
    const float bias = bd2[n];

    float* myTile = &sTile[waveInBlk][0];
#if HAVE_TDM
    const unsigned ldsOff = (unsigned)(unsigned long long)(void*)myTile;
#endif

    for (int t = tile0; t < numTiles; t += tileStride) {
        asm volatile("" ::: "memory");   // keep LDS operand reads inside the loop (no re-hoist/spill)

#if HAVE_TDM
        // stage this iteration's edge_attr 16x16 tile; overlaps with K-loop below
        tdm_load_tile16x16(ldsOff, edge_attr + (size_t)t * 16 * L + nt * 16, (unsigned)E);
#endif

        // ---- per-lane edge length (A-matrix row M = lane & 15) ----
        const int e = t * 16 + nl;
        const int r = edge_index[e];
        const int c = edge_index[E + e];
        const float* pr = mask[r] ? (last_pred + 3 * r) : (pos + 3 * r);
        const float* pc = mask[c] ? (last_pred + 3 * c) : (pos + 3 * c);
        const float dx = pr[0] - pc[0];
        const float dy = pr[1] - pc[1];
        const float dz = pr[2] - pc[2];
        const float len = sqrtf(dx * dx + dy * dy + dz * dz);

        // ---- 16x128 @ 128x16 via 32 x v_wmma_f32_16x16x4_f32 ----
        v8f acc = {0.f, 0.f, 0.f, 0.f, 0.f, 0.f, 0.f, 0.f};
#pragma unroll
        for (int kk = 0; kk < 32; ++kk) {
            const v4f cf = sC[(kk << 1) | hi];
            const v2f bf = sB[(kk << 5) + lane];
            v2f a;
            a.x = fmaxf(fmaf(len, cf.x, cf.z), 0.f);
            a.y = fmaxf(fmaf(len, cf.y, cf.w), 0.f);
            acc = __builtin_amdgcn_wmma_f32_16x16x4_f32(
                false, a, false, bf, (short)0, acc, false, false);
        }

#if HAVE_TDM
        __builtin_amdgcn_s_wait_tensorcnt(0);
#endif
#pragma unroll
        for (int rr = 0; rr < 8; ++rr) {
            const int    m   = rr + 8 * hi;            // C/D layout row
            const size_t idx = (size_t)(t * 16 + m) * L + n;
#if HAVE_TDM
            const float ea = myTile[m * 17 + nl];
#else
            const float ea = edge_attr[idx];
#endif
            out[idx] = acc[rr] + bias + ea;
        }
    }
}

// -------------------------------------------------------------------------
// Node path: out[v,:] = x[v,:] + relu(pos_m[v] @ Wp1 + bp1) @ Wp2 + bp2
// -------------------------------------------------------------------------
__global__ void node_mlp_wmma(
    const float* __restrict__ pos,
    const float* __restrict__ x,
    const float* __restrict__ last_pred,
    const unsigned char* __restrict__ mask,
    const float* __restrict__ Wp1,          // [3,L]
    const float* __restrict__ bp1,          // [L]
    const float* __restrict__ Wp2,          // [L,L]
    const float* __restrict__ bp2,          // [L]
    float* __restrict__ out,                // [N,L]
    int numTiles)
{
    __shared__ v2f sB[32 * 32];   // Wp2 col-block, fragment order
    __shared__ v4f sC0[64];       // {wx,wy,wz,b} for element k0
    __shared__ v4f sC1[64];       // {wx,wy,wz,b} for element k0+1

    const int nt = blockIdx.x & 7;

    for (int i = threadIdx.x; i < 32 * 32; i += blockDim.x) {
        const int lane_ = i & 31, kk_ = i >> 5;
        const int k0_ = kk_ * 4 + ((lane_ < 16) ? 0 : 2);
        const int n_  = nt * 16 + (lane_ & 15);
        v2f b; b.x = Wp2[k0_ * L + n_]; b.y = Wp2[(k0_ + 1) * L + n_];
        sB[i] = b;
    }
    for (int i = threadIdx.x; i < 64; i += blockDim.x) {
        const int k0_ = (i >> 1) * 4 + (i & 1) * 2;
        v4f c0; c0.x = Wp1[k0_];     c0.y = Wp1[L + k0_];     c0.z = Wp1[2 * L + k0_];     c0.w = bp1[k0_];
        v4f c1; c1.x = Wp1[k0_ + 1]; c1.y = Wp1[L + k0_ + 1]; c1.z = Wp1[2 * L + k0_ + 1]; c1.w = bp1[k0_ + 1];
        sC0[i] = c0;
        sC1[i] = c1;
    }
    __syncthreads();

    const int lane        = threadIdx.x & 31;
    const int waveInBlk   = threadIdx.x >> 5;
    const int wavesPerBlk = blockDim.x >> 5;
    const int tile0       = (blockIdx.x >> 3) * wavesPerBlk + waveInBlk;
    const int tileStride  = (gridDim.x >> 3) * wavesPerBlk;

    const int nl   = lane & 15;
    const int hi   = lane >> 4;
    const int n    = nt * 16 + nl;
    const float bias = bp2[n];

    for (int t = tile0; t < numTiles; t += tileStride) {
        asm volatile("" ::: "memory");

        const int v = t * 16 + nl;
        const float* pv = mask[v] ? (last_pred + 3 * v) : (pos + 3 * v);
        const float px = pv[0], py = pv[1], pz = pv[2];

        if (t + tileStride < numTiles) {
            __builtin_prefetch(&x[(size_t)(t + tileStride) * 16 * L + n], 0, 1);
        }

        v8f acc = {0.f, 0.f, 0.f, 0.f, 0.f, 0.f, 0.f, 0.f};
#pragma unroll
        for (int kk = 0; kk < 32; ++kk) {
            const int j = (kk << 1) | hi;
            const v4f c0 = sC0[j];
            const v4f c1 = sC1[j];
            const v2f bf = sB[(kk << 5) + lane];
            v2f a;
            a.x = fmaxf(fmaf(px, c0.x, fmaf(py, c0.y, fmaf(pz, c0.z, c0.w))), 0.f);
            a.y = fmaxf(fmaf(px, c1.x, fmaf(py, c1.y, fmaf(pz, c1.z, c1.w))), 0.f);
            acc = __builtin_amdgcn_wmma_f32_16x16x4_f32(
                false, a, false, bf, (short)0, acc, false, false);
        }

#pragma unroll
        for (int rr = 0; rr < 8; ++rr) {
            const size_t idx = (size_t)(t * 16 + rr + 8 * hi) * L + n;
            out[idx] = acc[rr] + bias + x[idx];
        }
    }
}

extern "C" void kernel_launch(void* const* d_in, const int* in_sizes, int n_in,
                              void* d_out, int out_size, void* d_ws, size_t ws_size,
                              hipStream_t stream) {
    const float*         pos        = (const float*)d_in[0];
    const float*         x          = (const float*)d_in[1];
    const float*         edge_attr  = (const float*)d_in[2];
    const int*           edge_index = (const int*)d_in[3];
    const float*         last_pred  = (const float*)d_in[4];
    const unsigned char* mask       = (const unsigned char*)d_in[5];
    const float* Wp1 = (const float*)d_in[6];
    const float* bp1 = (const float*)d_in[7];
    const float* Wp2 = (const float*)d_in[8];
    const float* bp2 = (const float*)d_in[9];
    const float* Wd1 = (const float*)d_in[10];
    const float* bd1 = (const float*)d_in[11];
    const float* Wd2 = (const float*)d_in[12];
    const float* bd2 = (const float*)d_in[13];

    const int N = in_sizes[0] / 3;   // 50000
    const int E = in_sizes[2] / L;   // 1600000

    float* out_x    = (float*)d_out;                  // [N,L]
    float* out_edge = (float*)d_out + (size_t)N * L;  // [E,L]

    const int nodeTiles = N / 16;    // 3125
    const int edgeTiles = E / 16;    // 100000

    // gridDim multiple of 8: low 3 bits of blockIdx select the feature column block
    node_mlp_wmma<<<128, 256, 0, stream>>>(
        pos, x, last_pred, mask, Wp1, bp1, Wp2, bp2, out_x, nodeTiles);

    edge_mlp_wmma<<<1024, 256, 0, stream>>>(
        pos, edge_index, edge_attr, last_pred, mask,
        Wd1, bd1, Wd2, bd2, out_edge, E, edgeTiles);
}